// LookupLayer_31911607009405
// MI455X (gfx1250) — compile-verified
//
#include <hip/hip_runtime.h>

// Embedding gather: out[row, :] = table[ids[row], :]
//   table: 1,000,000 x 64 f32   ids: 16384*50 int   out: 819200 x 64 f32
//
// Pure HBM-bound gather (0 FLOPs). Strategy:
//  - 16 lanes per row, one 128-bit chunk per lane: fully coalesced 256B row moves.
//  - 4 independent row-chunks per thread for memory-level parallelism.
//  - Table loads: default temporal (RT) -> ~143MB distinct working set stays in 192MB L2.
//  - Output stores: non-temporal (NT b128) -> 210MB write stream does not evict the table.
//  - global_prefetch_b8 (gfx1250) to pre-warm upcoming ids cachelines.

typedef __attribute__((ext_vector_type(4))) float v4f;  // native vector: OK for NT builtins

#define EMB      64
#define N_ROWS   (16384 * 50)          // 819,200 rows
#define N_CHUNKS (N_ROWS * 16)         // 13,107,200 float4 chunks
#define UNROLL   4
#define TPB      256

__global__ __launch_bounds__(TPB) void lookup_gather_kernel(
    const int*   __restrict__ ids,
    const float* __restrict__ table,
    float*       __restrict__ out)
{
    constexpr int T = N_CHUNKS / UNROLL;           // 3,276,800 threads
    const int t = blockIdx.x * TPB + threadIdx.x;  // grid sized exactly, no tail

    int idx[UNROLL];
    int row[UNROLL];
    int id [UNROLL];

    // Phase 1: load the 4 ids (independent, far-apart rows).
    #pragma unroll
    for (int k = 0; k < UNROLL; ++k) {
        idx[k] = t + k * T;
        row[k] = idx[k] >> 4;                      // row = chunk_index / 16
        id [k] = ids[row[k]];
    }

    // Pre-warm future ids cachelines (speculative, no counters, gfx1250 global_prefetch_b8).
    {
        int pf = row[0] + 4096;
        if (pf < N_ROWS) __builtin_prefetch(ids + pf, 0, 3);
    }

    // Phase 2: issue the 4 independent gather loads (b128 each; RT so table stays in L2).
    v4f v[UNROLL];
    #pragma unroll
    for (int k = 0; k < UNROLL; ++k) {
        const int chunk = idx[k] & 15;             // which 16B slice of the 256B row
        const v4f* src =
            (const v4f*)(table + (size_t)(unsigned)id[k] * EMB) + chunk;
        v[k] = *src;
    }

    // Phase 3: perfectly coalesced non-temporal b128 stores.
    // out float-offset of chunk idx is exactly idx*4 (since idx = row*16 + chunk).
    #pragma unroll
    for (int k = 0; k < UNROLL; ++k) {
        __builtin_nontemporal_store(v[k], (v4f*)out + idx[k]);
    }
}

extern "C" void kernel_launch(void* const* d_in, const int* in_sizes, int n_in,
                              void* d_out, int out_size, void* d_ws, size_t ws_size,
                              hipStream_t stream) {
    (void)in_sizes; (void)n_in; (void)out_size; (void)d_ws; (void)ws_size;

    const int*   ids   = (const int*)  d_in[0];
    const float* table = (const float*)d_in[1];
    float*       out   = (float*)      d_out;

    constexpr int total_threads = N_CHUNKS / UNROLL;   // 3,276,800
    constexpr int blocks        = total_threads / TPB; // 12,800 (exact)

    lookup_gather_kernel<<<blocks, TPB, 0, stream>>>(ids, table, out);
}